// LlamaRMSNormInt4_61856118997383
// MI455X (gfx1250) — compile-verified
//
#include <hip/hip_runtime.h>

#ifndef __has_builtin
#define __has_builtin(x) 0
#endif

#define TPB 256
#define HDIM 4096
#define NWAVES (TPB / 32)

// Device pass: builtin exists -> async path. Host pass: builtin unknown ->
// fallback path is parsed (but never executed on host).
#if __has_builtin(__builtin_amdgcn_global_load_async_to_lds_b128)
#define HAVE_ASYNC_LDS 1
#else
#define HAVE_ASYNC_LDS 0
#endif

#if __has_builtin(__builtin_amdgcn_s_wait_asynccnt)
#define ASYNC_WAIT(n) __builtin_amdgcn_s_wait_asynccnt(n)
#elif HAVE_ASYNC_LDS
#define ASYNC_WAIT(n) asm volatile("s_wait_asynccnt %0" ::"n"(n) : "memory")
#else
#define ASYNC_WAIT(n) ((void)0)
#endif

// Match the builtin's parameter types exactly: GCC-style int 4-vector,
// AS1 (global, prints as __device__) and AS3 (LDS, prints as __shared__).
typedef int v4i __attribute__((vector_size(16)));
typedef __attribute__((address_space(1))) v4i gv4i;
typedef __attribute__((address_space(3))) v4i lv4i;

// Stage one 16 KB span into LDS: 1024 x b128; each of 256 threads moves 4
// 16-byte chunks => 4 async-to-LDS b128 ops per wave (ASYNCcnt += 4 per wave).
__device__ __forceinline__ void issue_span16k(const void* gsrc, void* ldst,
                                              int tid) {
#if HAVE_ASYNC_LDS
#pragma unroll
  for (int k = 0; k < 4; ++k) {
    const char* g = (const char*)gsrc + (size_t)tid * 16 + (size_t)k * 4096;
    char*       l = (char*)ldst + tid * 16 + k * 4096;
    __builtin_amdgcn_global_load_async_to_lds_b128(
        (gv4i*)(void*)g, (lv4i*)(void*)l, 0, 0);
  }
#else
  const v4i* g4 = (const v4i*)gsrc;
  v4i*       l4 = (v4i*)ldst;
#pragma unroll
  for (int k = 0; k < 4; ++k) l4[tid + k * 256] = g4[tid + k * 256];
#endif
}

__global__ __launch_bounds__(TPB) void rmsnorm_reorder_int4_kernel(
    const float* __restrict__ x, const float* __restrict__ w,
    const int* __restrict__ idx, float* __restrict__ qout,
    float* __restrict__ sout, int rows, int rows_per_block) {
  __shared__ float w_lds[HDIM];      // original-order weight (pass 1)
  __shared__ int   idx_lds[HDIM];    // permutation
  __shared__ float wperm_lds[HDIM];  // w[idx[j]] (row-invariant, pass 2)
  __shared__ float x_lds[2][HDIM];   // double-buffered row
  __shared__ float red_ss[NWAVES];
  __shared__ float red_mx[NWAVES];
  __shared__ float s_rinv, s_rs;

  const int tid = threadIdx.x;

  int row0   = (int)blockIdx.x * rows_per_block;
  int rowEnd = row0 + rows_per_block;
  if (rowEnd > rows) rowEnd = rows;
  if (row0 >= rowEnd) return;

  // Prologue: async-stage w, idx, then first two rows (16 b128 ops/wave).
  issue_span16k(w, w_lds, tid);
  issue_span16k(idx, idx_lds, tid);
  issue_span16k(x + (size_t)row0 * HDIM, x_lds[0], tid);
  if (row0 + 1 < rowEnd)
    issue_span16k(x + (size_t)(row0 + 1) * HDIM, x_lds[1], tid);

  // ASYNCcnt retires in order: <=8 outstanding => w and idx have landed.
  ASYNC_WAIT(8);
  __syncthreads();

  // Row-invariant permuted weight: one coalesced read per element in pass 2
  // instead of a random gather per row. (Visible after the loop's barrier.)
  for (int j = tid; j < HDIM; j += TPB) wperm_lds[j] = w_lds[idx_lds[j]];

  for (int r = row0; r < rowEnd; ++r) {
    const int buf = (r - row0) & 1;

    // Current row resident; next row's 4 ops may stay in flight.
    if (r + 1 < rowEnd) { ASYNC_WAIT(4); } else { ASYNC_WAIT(0); }
    __syncthreads();

    // Pass 1: sum(x^2) and max|x*w| in one sweep over LDS (float4 strided).
    float ss = 0.0f, mx = 0.0f;
    const float4* x4 = (const float4*)x_lds[buf];
    const float4* w4 = (const float4*)w_lds;
#pragma unroll
    for (int k = 0; k < HDIM / 4 / TPB; ++k) {
      int    j  = tid + k * TPB;
      float4 xv = x4[j];
      float4 wv = w4[j];
      ss += xv.x * xv.x + xv.y * xv.y + xv.z * xv.z + xv.w * xv.w;
      float m0 = fmaxf(fabsf(xv.x * wv.x), fabsf(xv.y * wv.y));
      float m1 = fmaxf(fabsf(xv.z * wv.z), fabsf(xv.w * wv.w));
      mx = fmaxf(mx, fmaxf(m0, m1));
    }
    // wave32 reduction
#pragma unroll
    for (int off = 16; off > 0; off >>= 1) {
      ss += __shfl_xor(ss, off, 32);
      mx = fmaxf(mx, __shfl_xor(mx, off, 32));
    }
    const int wid = tid >> 5;
    if ((tid & 31) == 0) { red_ss[wid] = ss; red_mx[wid] = mx; }
    __syncthreads();

    if (tid == 0) {
      float S = 0.0f, M = 0.0f;
#pragma unroll
      for (int i = 0; i < NWAVES; ++i) {
        S += red_ss[i];
        M = fmaxf(M, red_mx[i]);
      }
      float rinv  = rsqrtf(S * (1.0f / (float)HDIM) + 1e-6f);
      // scale is permutation-invariant: max|y| = rinv * max|x*w|
      float scale = fmaxf(M * rinv * (1.0f / 7.0f), 1e-8f);
      s_rinv  = rinv;
      s_rs    = rinv / scale;  // combined normalize+quantize multiplier
      sout[r] = scale;
    }
    __syncthreads();

    const float rs = s_rs;
    float* __restrict__ qrow = qout + (size_t)r * HDIM;

    // Pass 2: permuted gather from LDS + quantize; vectorized index/weight
    // loads and b128 coalesced global stores.
    const int4*   idx4 = (const int4*)idx_lds;
    const float4* wp4  = (const float4*)wperm_lds;
    const float*  xb   = x_lds[buf];
    float4*       q4   = (float4*)qrow;
#pragma unroll
    for (int k = 0; k < HDIM / 4 / TPB; ++k) {
      int    j = tid + k * TPB;
      int4   c = idx4[j];
      float4 wv = wp4[j];
      float4 q;
      q.x = fminf(fmaxf(rintf(xb[c.x] * wv.x * rs), -8.0f), 7.0f);
      q.y = fminf(fmaxf(rintf(xb[c.y] * wv.y * rs), -8.0f), 7.0f);
      q.z = fminf(fmaxf(rintf(xb[c.z] * wv.z * rs), -8.0f), 7.0f);
      q.w = fminf(fmaxf(rintf(xb[c.w] * wv.w * rs), -8.0f), 7.0f);
      q4[j] = q;
    }
    __syncthreads();  // everyone done reading buf before refilling it

    if (r + 2 < rowEnd) issue_span16k(x + (size_t)(r + 2) * HDIM, x_lds[buf], tid);
  }
}

extern "C" void kernel_launch(void* const* d_in, const int* in_sizes, int n_in,
                              void* d_out, int out_size, void* d_ws,
                              size_t ws_size, hipStream_t stream) {
  (void)n_in; (void)d_ws; (void)ws_size; (void)out_size;
  const float* x   = (const float*)d_in[0];
  const float* w   = (const float*)d_in[1];
  const int*   idx = (const int*)d_in[2];

  const int H     = in_sizes[1];          // 4096
  const int total = in_sizes[0];          // B*S*H
  const int rows  = total / H;            // B*S tokens

  float* qout = (float*)d_out;            // q values (int4 range) as float
  float* sout = (float*)d_out + (size_t)total;  // per-token scales

  int grid = 1024;
  if (grid > rows) grid = rows;
  int rpb = (rows + grid - 1) / grid;

  hipLaunchKernelGGL(rmsnorm_reorder_int4_kernel, dim3(grid), dim3(TPB), 0,
                     stream, x, w, idx, qout, sout, rows, rpb);
}